// Akima_66623532696299
// MI455X (gfx1250) — compile-verified
//
#include <hip/hip_runtime.h>
#include <hip/hip_bf16.h>

// Akima spline evaluation, fp32, memory-bound streaming kernel for gfx1250.
//
// Strategy:
//  - Each block rebuilds the 63-interval coefficient table (c0,c1,c2,c3) in
//    LDS from the 64-node `value` array (cost ~O(64), negligible vs the
//    ~1K float4 elements each block streams).
//  - Table fill uses GLOBAL_LOAD_ASYNC_TO_LDS_B32 + s_wait_asynccnt when the
//    toolchain exposes the builtin (CDNA5 async path), else a plain LDS fill.
//  - Main loop: grid-stride over float4 chunks, non-temporal b128 loads and
//    stores (data is touched exactly once; 403MB > 192MB L2, so NT avoids
//    L2 thrashing), per-element ds_load_b128 coefficient fetch, 3-FMA Horner.

typedef float v4f __attribute__((ext_vector_type(4)));

typedef __attribute__((address_space(1))) int g_int;
typedef __attribute__((address_space(3))) int l_int;

#ifndef MAX_NODES
#define MAX_NODES 256
#endif

__global__ void __launch_bounds__(256)
akima_eval_kernel(const float* __restrict__ x,
                  const float* __restrict__ value,
                  float* __restrict__ out,
                  int n, int nv)
{
    __shared__ float sval[MAX_NODES];        // node values
    __shared__ float m_ext[MAX_NODES + 3];   // extended slopes (nv+3 used)
    __shared__ float tt[MAX_NODES];          // Akima tangents
    __shared__ v4f   coef[MAX_NODES - 1];    // per-interval {c0,c1,c2,c3}

    const int tid = threadIdx.x;
    const float h     = 1.0f / (float)(nv - 1);
    const float inv_h = (float)(nv - 1);

    // ---- load node values into LDS (async-to-LDS path if available) ----
#if defined(__has_builtin) && __has_builtin(__builtin_amdgcn_global_load_async_to_lds_b32)
    if (tid < nv) {
        __builtin_amdgcn_global_load_async_to_lds_b32(
            (g_int*)(value + tid),
            (l_int*)(&sval[tid]),
            0, 0);
    }
#if defined(__has_builtin) && __has_builtin(__builtin_amdgcn_s_wait_asynccnt)
    __builtin_amdgcn_s_wait_asynccnt(0);
#else
    asm volatile("s_wait_asynccnt 0" ::: "memory");
#endif
#else
    if (tid < nv) sval[tid] = value[tid];
#endif
    __syncthreads();

    // ---- slopes m[i] = diff(value)/h into m_ext[2..nv] ----
    if (tid < nv - 1) {
        m_ext[tid + 2] = (sval[tid + 1] - sval[tid]) / h;
    }
    __syncthreads();

    // ---- extend slopes at both ends (Akima extrapolation) ----
    if (tid == 0) {
        const float m0 = m_ext[2], m1 = m_ext[3];
        m_ext[1] = 2.0f * m0 - m1;
        m_ext[0] = 3.0f * m0 - 2.0f * m1;
        const float ml1 = m_ext[nv];       // m[-1]
        const float ml2 = m_ext[nv - 1];   // m[-2]
        m_ext[nv + 1] = 2.0f * ml1 - ml2;
        m_ext[nv + 2] = 3.0f * ml1 - 2.0f * ml2;
    }
    __syncthreads();

    // ---- Akima tangents t[i], i in [0, nv) ----
    if (tid < nv) {
        const float w1 = fabsf(m_ext[tid + 3] - m_ext[tid + 2]); // dm[i+2]
        const float w2 = fabsf(m_ext[tid + 1] - m_ext[tid]);     // dm[i]
        const float ml = m_ext[tid + 1];
        const float mr = m_ext[tid + 2];
        const float denom = w1 + w2;
        tt[tid] = (denom > 0.0f) ? (w1 * ml + w2 * mr) / denom
                                 : 0.5f * (ml + mr);
    }
    __syncthreads();

    // ---- cubic coefficients per interval ----
    if (tid < nv - 1) {
        const float mi = m_ext[tid + 2];
        const float t0 = tt[tid];
        const float t1 = tt[tid + 1];
        v4f c;
        c.x = sval[tid];
        c.y = t0;
        c.z = (3.0f * mi - 2.0f * t0 - t1) / h;
        c.w = (t0 + t1 - 2.0f * mi) / (h * h);
        coef[tid] = c;
    }
    __syncthreads();

    // ---- main streaming loop: float4 grid-stride, NT loads/stores ----
    const int gtid    = blockIdx.x * blockDim.x + threadIdx.x;
    const int gstride = gridDim.x * blockDim.x;
    const int total4  = n >> 2;
    const int idx_max = nv - 2;

    const v4f* __restrict__ in4  = (const v4f*)x;
    v4f* __restrict__       out4 = (v4f*)out;

    for (int i = gtid; i < total4; i += gstride) {
        const v4f xv = __builtin_nontemporal_load(in4 + i);
        v4f r;
        #pragma unroll
        for (int k = 0; k < 4; ++k) {
            float xe = (k == 0) ? xv.x : (k == 1) ? xv.y : (k == 2) ? xv.z : xv.w;
            float xc = fminf(fmaxf(xe, 0.0f), 1.0f);
            int idx = (int)floorf(xc * inv_h);
            idx = idx < 0 ? 0 : (idx > idx_max ? idx_max : idx);
            const float s = fmaf(-(float)idx, h, xc);
            const v4f c = coef[idx];                 // ds_load_b128
            float acc = fmaf(s, c.w, c.z);
            acc = fmaf(s, acc, c.y);
            acc = fmaf(s, acc, c.x);
            if (k == 0) r.x = acc; else if (k == 1) r.y = acc;
            else if (k == 2) r.z = acc; else r.w = acc;
        }
        __builtin_nontemporal_store(r, out4 + i);
    }

    // ---- scalar tail (n not divisible by 4) ----
    const int rem_start = total4 << 2;
    for (int j = rem_start + gtid; j < n; j += gstride) {
        float xc = fminf(fmaxf(x[j], 0.0f), 1.0f);
        int idx = (int)floorf(xc * inv_h);
        idx = idx < 0 ? 0 : (idx > idx_max ? idx_max : idx);
        const float s = fmaf(-(float)idx, h, xc);
        const v4f c = coef[idx];
        float acc = fmaf(s, c.w, c.z);
        acc = fmaf(s, acc, c.y);
        acc = fmaf(s, acc, c.x);
        out[j] = acc;
    }
}

extern "C" void kernel_launch(void* const* d_in, const int* in_sizes, int n_in,
                              void* d_out, int out_size, void* d_ws, size_t ws_size,
                              hipStream_t stream) {
    (void)n_in; (void)d_ws; (void)ws_size; (void)out_size;

    const float* x     = (const float*)d_in[0];
    const float* value = (const float*)d_in[1];
    float* out         = (float*)d_out;

    const int n  = in_sizes[0];   // 64*3*512*512 = 50,331,648
    const int nv = in_sizes[1];   // 64 nodes

    const int threads = 256;                    // 8 wave32 waves per block
    const int iters   = 4;                      // ~4 float4 per thread
    long long total4  = (long long)(n >> 2);
    long long b = (total4 + (long long)threads * iters - 1) /
                  ((long long)threads * iters);
    if (b < 1) b = 1;
    if (b > 1048576) b = 1048576;
    const int blocks = (int)b;

    akima_eval_kernel<<<blocks, threads, 0, stream>>>(x, value, out, n, nv);
}